// COFunc_9105330668116
// MI455X (gfx1250) — compile-verified
//
#include <hip/hip_runtime.h>

// Problem shape (fixed by the reference): B=2, N=50000, P=64, E=800000.
// out[b,n,0:P)  = dv  (atomic-accumulated, 1/m folded into edge weights)
// out[b,n,P:2P) = dr  = v = u[b,n,0:P)
static constexpr int P_    = 64;        // half feature dim
static constexpr int ROW   = 2 * P_;    // 128 floats per (b,n) row
static constexpr int BLOCK = 256;       // 8 wave32s per workgroup

// Native clang vector type: accepted by __builtin_nontemporal_* and lowers to
// b128 vector memory ops.
typedef float v4f __attribute__((ext_vector_type(4)));

// ---------------------------------------------------------------------------
// Kernel A: zero the dv region, stream-copy v -> dr.
// One thread per float4 of the P-block => B*N*P/4 threads.
// dr is write-once / never re-read on device -> non-temporal load+store so the
// 25.6MB stream does not evict r/dv lines from L2 (the whole ~51MB hot set
// fits in MI455X's 192MB L2; keep it resident for the gather/atomic kernel).
// ---------------------------------------------------------------------------
__global__ void init_out_kernel(const v4f* __restrict__ u4,
                                v4f* __restrict__ out4,
                                int total4)
{
    int idx = blockIdx.x * blockDim.x + threadIdx.x;
    if (idx >= total4) return;
    size_t row = (size_t)(idx >> 4);   // (b*N + n)
    int    c   = idx & 15;             // float4 chunk within the P block

    // dv = 0 (regular-temporal store: these lines are about to be hammered
    // by L2 atomics, keep them cached)
    out4[row * 32 + c] = (v4f){0.f, 0.f, 0.f, 0.f};

    // dr = v (non-temporal both ways: v read exactly once, dr written once)
    v4f v = __builtin_nontemporal_load(&u4[row * 32 + c]);
    __builtin_nontemporal_store(v, &out4[row * 32 + 16 + c]);
}

// ---------------------------------------------------------------------------
// Kernel B: one wave32 per undirected edge (wave id made explicitly scalar so
// edge metadata goes down the SMEM/SALU path instead of redundant per-lane
// vector loads).
//   lane>>4  -> batch b in {0,1}
//   lane&15  -> float4 chunk c in [0,16) of the P=64 features
// Each half-wave's 16 lanes cover one contiguous 256B node row -> coalesced
// global_load_b128 gathers (L2 hits after warmup). Contributions:
//   node a += (k/m[a]) * (r_b - r_a)
//   node b -= (k/m[b]) * (r_b - r_a)
// via non-returning global_atomic_add_f32 (unsafeAtomicAdd).
// ---------------------------------------------------------------------------
__global__ __launch_bounds__(BLOCK)
void edge_scatter_kernel(const float* __restrict__ u,
                         const int*   __restrict__ ei,   // [2*E]
                         const float* __restrict__ kw,   // [E]
                         const float* __restrict__ m,    // [N]
                         float* __restrict__ out,
                         int E, int N)
{
    // Wave-uniform edge index: readfirstlane makes uniformity provable, so
    // a/b/w/m loads below become scalar (s_load) and their math SALU.
    int wid = __builtin_amdgcn_readfirstlane((int)(threadIdx.x >> 5));
    int e   = blockIdx.x * (BLOCK / 32) + wid;
    if (e >= E) return;
    int lane = threadIdx.x & 31;

    // Prefetch edge metadata 64 edges (8 blocks) ahead (gfx1250 global_prefetch_b8).
    if (lane == 0 && (e + 64) < E) {
        __builtin_prefetch(&ei[e + 64],     0, 0);
        __builtin_prefetch(&ei[E + e + 64], 0, 0);
        __builtin_prefetch(&kw[e + 64],     0, 0);
    }

    int   a = ei[e];
    int   b = ei[E + e];
    float w = kw[e];
    // Fold segment_sum's final 1/m into the edge weight. v_rcp_f32 (~1ulp,
    // exact for m==1) instead of a per-lane division macro.
    float wa = w * __builtin_amdgcn_rcpf(m[a]);
    float wb = w * __builtin_amdgcn_rcpf(m[b]);

    int bb = lane >> 4;    // batch index
    int c  = lane & 15;    // float4 chunk within P

    size_t rowA = ((size_t)bb * (size_t)N + (size_t)a) * ROW;
    size_t rowB = ((size_t)bb * (size_t)N + (size_t)b) * ROW;

    v4f ra = *(const v4f*)(u + rowA + P_ + c * 4);  // r[bb, a, c*4..]
    v4f rb = *(const v4f*)(u + rowB + P_ + c * 4);  // r[bb, b, c*4..]

    v4f d = rb - ra;

    float* oa = out + rowA + c * 4;   // dv region of node a
    float* ob = out + rowB + c * 4;   // dv region of node b

    unsafeAtomicAdd(oa + 0,  wa * d.x);
    unsafeAtomicAdd(oa + 1,  wa * d.y);
    unsafeAtomicAdd(oa + 2,  wa * d.z);
    unsafeAtomicAdd(oa + 3,  wa * d.w);
    unsafeAtomicAdd(ob + 0, -wb * d.x);
    unsafeAtomicAdd(ob + 1, -wb * d.y);
    unsafeAtomicAdd(ob + 2, -wb * d.z);
    unsafeAtomicAdd(ob + 3, -wb * d.w);
}

// ---------------------------------------------------------------------------
// Launch. Input order (setup_inputs dict): u, k, m, edge_index, t.
// ---------------------------------------------------------------------------
extern "C" void kernel_launch(void* const* d_in, const int* in_sizes, int n_in,
                              void* d_out, int out_size, void* d_ws, size_t ws_size,
                              hipStream_t stream)
{
    const float* u  = (const float*)d_in[0];
    const float* kw = (const float*)d_in[1];
    const float* m  = (const float*)d_in[2];
    const int*   ei = (const int*)  d_in[3];
    float*       out = (float*)d_out;

    const int E = in_sizes[1];           // 800000
    const int N = in_sizes[2];           // 50000

    // total float4s in the dv (== dr) half: B*N*P/4 = out_size / 8
    const int total4 = out_size / 8;

    {
        dim3 block(BLOCK);
        dim3 grid((total4 + BLOCK - 1) / BLOCK);
        init_out_kernel<<<grid, block, 0, stream>>>((const v4f*)u,
                                                    (v4f*)out, total4);
    }
    {
        // one wave32 per edge, 8 edges per block
        const int edges_per_block = BLOCK / 32;
        dim3 block(BLOCK);
        dim3 grid((E + edges_per_block - 1) / edges_per_block);
        edge_scatter_kernel<<<grid, block, 0, stream>>>(u, ei, kw, m, out, E, N);
    }
}